// GCN_79285096284452
// MI455X (gfx1250) — compile-verified
//
#include <hip/hip_runtime.h>
#include <hip/hip_bf16.h>

// GCN forward for MI455X (gfx1250), wave32 / WMMA.
//   h = tanh(x @ encW + encb)
//   3x: hw = h @ Wi ; agg = bi + scatter_add(hw[src]*w, dst) ; h = tanh(agg)
//   out = h @ decW + decb
// GEMMs use V_WMMA_F32_16X16X4_F32 (full f32 fidelity; GEMM is not the
// bottleneck — the L2-resident edge scatter-add is). tanh(agg) is fused into
// the NEXT gemm's A-tile LDS load (PRE_TANH) to avoid 3 extra N*H passes.

typedef float v2f __attribute__((ext_vector_type(2)));
typedef float v8f __attribute__((ext_vector_type(8)));

__device__ __forceinline__ float gpu_tanh(float x) {
  float y;
  asm("v_tanh_f32 %0, %1" : "=v"(y) : "v"(x));
  return y;
}

// ---------------------------------------------------------------------------
// GEMM: C[M x NC] = act( A[M x 128] @ B[128 x NC] + bias )
//   - 256 threads = 8 waves per block; block covers 64 rows.
//   - A tile staged in LDS (row stride 132 dwords -> conflict-free b64 reads).
//   - Each wave owns one 16-col tile (constant ct) and 4/2 row tiles, so the
//     B fragment is loaded once per k-step and reused across all row tiles.
//   - PRE_TANH applies tanh elementwise while filling LDS (fuses prior layer's
//     activation); POST_TANH/HAS_BIAS apply in the epilogue.
// ---------------------------------------------------------------------------
template <int NC, bool PRE_TANH, bool POST_TANH, bool HAS_BIAS>
__global__ __launch_bounds__(256) void gemm128_wmma(
    const float* __restrict__ A, const float* __restrict__ B,
    const float* __restrict__ bias, float* __restrict__ C, int M) {
  constexpr int K = 128;
  constexpr int MB = 64;          // rows per block
  constexpr int LDA = K + 4;      // 132 dwords: 132 % 64 == 4 -> no bank conflicts
  __shared__ float As[MB * LDA];

  const int tid = threadIdx.x;
  const int wave = tid >> 5;
  const int lane = tid & 31;
  const int hi = lane >> 4;       // lane half: selects K pair / M+8
  const int lr = lane & 15;
  const int rowBase = blockIdx.x * MB;

  // ---- stage 64x128 A tile into LDS (zero-pad OOB rows, optional tanh) ----
  {
#pragma unroll
    for (int i = 0; i < 8; ++i) {
      const int idx = tid + i * 256;        // 2048 float4 slots
      const int row = idx >> 5;
      const int c4 = (idx & 31) << 2;
      const int grow = rowBase + row;
      float4 v = make_float4(0.f, 0.f, 0.f, 0.f);
      if (grow < M) {
        v = *(const float4*)(A + (size_t)grow * K + c4);
        if (PRE_TANH) {
          v.x = gpu_tanh(v.x); v.y = gpu_tanh(v.y);
          v.z = gpu_tanh(v.z); v.w = gpu_tanh(v.w);
        }
      }
      *(float4*)(&As[row * LDA + c4]) = v;  // row*132 floats = 528B, 16B aligned
    }
  }
  __syncthreads();

  constexpr int NCT = NC / 16;              // col tiles in block
  constexpr int NT = (4 * NCT) / 8;         // tiles per wave (4 row tiles/block)
  constexpr int RSTEP = 8 / NCT;

  const int ctile = wave % NCT;             // constant per wave
  const int rbase = wave / NCT;
  const int bcol = ctile * 16 + lr;

  v8f acc[NT];
#pragma unroll
  for (int t = 0; t < NT; ++t) acc[t] = (v8f)(0.f);

  const float* __restrict__ Bp = B + bcol;

  for (int kk = 0; kk < K; kk += 4) {
    const int k0 = kk + hi * 2;             // lanes 0-15: K0,K1 ; 16-31: K2,K3
    v2f bfrag;
    bfrag.x = Bp[(size_t)(k0) * NC];
    bfrag.y = Bp[(size_t)(k0 + 1) * NC];
#pragma unroll
    for (int t = 0; t < NT; ++t) {
      const int rt = rbase + RSTEP * t;
      const v2f afrag = *(const v2f*)(&As[(rt * 16 + lr) * LDA + k0]);
      acc[t] = __builtin_amdgcn_wmma_f32_16x16x4_f32(
          /*neg_a=*/false, afrag, /*neg_b=*/false, bfrag,
          /*c_mod=*/(short)0, acc[t], /*reuse_a=*/false, /*reuse_b=*/false);
    }
  }

  // ---- epilogue: bias / tanh / bounded store ----
  float bv = 0.f;
  if constexpr (HAS_BIAS) bv = bias[bcol];
#pragma unroll
  for (int t = 0; t < NT; ++t) {
    const int rt = rbase + RSTEP * t;
#pragma unroll
    for (int r = 0; r < 8; ++r) {           // VGPR r -> row r (+8 for hi half)
      const int row = rowBase + rt * 16 + r + hi * 8;
      if (row < M) {
        float v = acc[t][r] + bv;
        if (POST_TANH) v = gpu_tanh(v);
        C[(size_t)row * NC + bcol] = v;
      }
    }
  }
}

// ---------------------------------------------------------------------------
// agg[n][0..127] = bias[0..127]   (bias broadcast init before scatter-add)
// ---------------------------------------------------------------------------
__global__ __launch_bounds__(256) void init_bias_kernel(
    float* __restrict__ agg, const float* __restrict__ bias, int M) {
  const int idx = blockIdx.x * 256 + threadIdx.x;   // one float4 per thread
  const int total = M * 32;
  if (idx >= total) return;
  const int c4 = (idx & 31) << 2;
  ((float4*)agg)[idx] = *(const float4*)(bias + c4);
}

// ---------------------------------------------------------------------------
// Edge scatter: agg[dst[e]] += hw[src[e]] * w[e]
// One wave per edge: lane holds float4 -> one 512B row per wave, fully
// coalesced gather (L2-resident), 4 global_atomic_add_f32 per lane.
// ---------------------------------------------------------------------------
__global__ __launch_bounds__(256) void edge_scatter_kernel(
    const float* __restrict__ hw, const int* __restrict__ src,
    const int* __restrict__ dst, const float* __restrict__ w,
    float* __restrict__ agg, int E) {
  const int gwave = (blockIdx.x * 256 + threadIdx.x) >> 5;
  const int lane = threadIdx.x & 31;
  if (gwave >= E) return;                   // wave-uniform exit
  const int s = src[gwave];
  const int d = dst[gwave];
  const float ww = w[gwave];
  const float4 v = *(const float4*)(hw + (size_t)s * 128 + (lane << 2));
  float* ap = agg + (size_t)d * 128 + (lane << 2);
  atomicAdd(ap + 0, v.x * ww);
  atomicAdd(ap + 1, v.y * ww);
  atomicAdd(ap + 2, v.z * ww);
  atomicAdd(ap + 3, v.w * ww);
}

// ---------------------------------------------------------------------------
extern "C" void kernel_launch(void* const* d_in, const int* in_sizes, int n_in,
                              void* d_out, int out_size, void* d_ws,
                              size_t ws_size, hipStream_t stream) {
  const float* x     = (const float*)d_in[0];   // [N,128]
  const int*   eidx  = (const int*)d_in[1];     // [2,E]
  const float* ew    = (const float*)d_in[2];   // [E]
  const float* encW  = (const float*)d_in[3];   // [128,128]
  const float* encb  = (const float*)d_in[4];   // [128]
  const float* convW = (const float*)d_in[5];   // [3,128,128]
  const float* convb = (const float*)d_in[6];   // [3,128]
  const float* decW  = (const float*)d_in[7];   // [128,64]
  const float* decb  = (const float*)d_in[8];   // [64]
  float* out = (float*)d_out;

  const int M = in_sizes[0] / 128;              // nodes
  const int E = in_sizes[2];                    // edges
  const int* src = eidx;
  const int* dst = eidx + E;

  const size_t slab = (size_t)M * 128;          // floats per [N,H] buffer
  if (ws_size < 3 * slab * sizeof(float)) return;
  float* s0 = (float*)d_ws;
  float* s1 = s0 + slab;
  float* s2 = s1 + slab;

  const int gemmBlocks = (M + 63) / 64;
  const int initBlocks = (M * 32 + 255) / 256;
  const int edgeBlocks = (E + 7) / 8;           // 8 waves (edges) per block

  // Encoder: s0 = tanh(x @ encW + encb)
  gemm128_wmma<128, false, true, true>
      <<<gemmBlocks, 256, 0, stream>>>(x, encW, encb, s0, M);

  // Layer 0: hw=s1, agg=s2
  gemm128_wmma<128, false, false, false>
      <<<gemmBlocks, 256, 0, stream>>>(s0, convW, nullptr, s1, M);
  init_bias_kernel<<<initBlocks, 256, 0, stream>>>(s2, convb, M);
  edge_scatter_kernel<<<edgeBlocks, 256, 0, stream>>>(s1, src, dst, ew, s2, E);

  // Layer 1: A=s2 (tanh fused), hw=s0, agg=s1
  gemm128_wmma<128, true, false, false>
      <<<gemmBlocks, 256, 0, stream>>>(s2, convW + 128 * 128, nullptr, s0, M);
  init_bias_kernel<<<initBlocks, 256, 0, stream>>>(s1, convb + 128, M);
  edge_scatter_kernel<<<edgeBlocks, 256, 0, stream>>>(s0, src, dst, ew, s1, E);

  // Layer 2: A=s1 (tanh fused), hw=s2, agg=s0
  gemm128_wmma<128, true, false, false>
      <<<gemmBlocks, 256, 0, stream>>>(s1, convW + 2 * 128 * 128, nullptr, s2, M);
  init_bias_kernel<<<initBlocks, 256, 0, stream>>>(s0, convb + 256, M);
  edge_scatter_kernel<<<edgeBlocks, 256, 0, stream>>>(s2, src, dst, ew, s0, E);

  // Decoder: out = tanh-fused(s0) @ decW + decb   [N,64]
  gemm128_wmma<64, true, false, true>
      <<<gemmBlocks, 256, 0, stream>>>(s0, decW, decb, out, M);
}